// GCNClassifier_30751965839771
// MI455X (gfx1250) — compile-verified
//
#include <hip/hip_runtime.h>
#include <math.h>

typedef __attribute__((ext_vector_type(2))) float v2f;
typedef __attribute__((ext_vector_type(8))) float v8f;

#define IN_CH 128
#define HID 64
#define NC 10
#define NG 64

// ---------------- degree / normalization ----------------

__global__ void k_init_deg(float* deg, int N) {
    int i = blockIdx.x * blockDim.x + threadIdx.x;
    if (i < N) deg[i] = 1.0f;  // self-loop contributes 1
}

__global__ void k_edge_deg(const int* __restrict__ dst, float* deg, int E) {
    int e = blockIdx.x * blockDim.x + threadIdx.x;
    if (e < E) atomicAdd(&deg[dst[e]], 1.0f);
}

__global__ void k_finalize_dinv(float* deg, int N) {
    int i = blockIdx.x * blockDim.x + threadIdx.x;
    if (i < N) {
        float d = deg[i];
        deg[i] = (d > 0.0f) ? rsqrtf(d) : 0.0f;
    }
}

// ---------------- WMMA fp32 GEMM: H[N,C] = act(X[N,K]) @ W[K,C] ----------------
// One wave32 owns a 16-row tile and ALL NT column tiles. Per k-step:
//   - A fragment (global_load_b64) is software-pipelined one step ahead,
//   - all NT B fragments are loaded into distinct registers (batched ds loads,
//     single dscnt wait), then NT v_wmma_f32_16x16x4_f32 issue back-to-back.
// W is staged transposed in LDS (sW[col*K + k]); columns >= C are clamped, not
// masked — they only feed output columns that are never stored.
template <int NT, bool RELU>
__global__ void k_gemm_wmma(const float* __restrict__ X, const float* __restrict__ W,
                            float* __restrict__ H, int N, int K, int C, int ldh) {
    extern __shared__ float sW[];  // [NT*16][K], transposed
    const int tid = threadIdx.x;
    const int CP = NT * 16;

    for (int idx = tid; idx < CP * K; idx += blockDim.x) {
        int c = idx / K;
        int k = idx - c * K;
        int cc = (c < C) ? c : (C - 1);  // clamp padded columns
        sW[idx] = W[(size_t)k * C + cc];
    }
    __syncthreads();

    const int wave = (blockIdx.x * blockDim.x + tid) >> 5;
    const int lane = tid & 31;
    const int tilesM = (N + 15) >> 4;
    if (wave >= tilesM) return;  // wave-uniform exit

    const int half = lane >> 4;  // K sub-pair select (0: k..k+1, 1: k+2..k+3)
    const int l = lane & 15;

    int arow = wave * 16 + l;
    if (arow >= N) arow = N - 1;  // N=100000 is 16-aligned; safety clamp
    const float* __restrict__ xrow = X + (size_t)arow * K + half * 2;
    const float* __restrict__ brow = sW + (size_t)l * K + half * 2;

    v8f acc[NT];
#pragma unroll
    for (int t = 0; t < NT; ++t) acc[t] = (v8f){};

    // prologue: first A fragment in flight
    v2f a;
    a.x = xrow[0];
    a.y = xrow[1];

    for (int k = 0; k < K - 4; k += 4) {
        // prefetch next A fragment (hides global latency under the WMMAs)
        v2f an;
        an.x = xrow[k + 4];
        an.y = xrow[k + 5];

        v2f av = a;
        if (RELU) { av.x = fmaxf(av.x, 0.0f); av.y = fmaxf(av.y, 0.0f); }

        // batch all B fragment loads -> one dscnt wait, then NT back-to-back WMMAs
        v2f b[NT];
#pragma unroll
        for (int t = 0; t < NT; ++t) {
            const float* bp = brow + (size_t)t * 16 * K + k;
            b[t].x = bp[0];
            b[t].y = bp[1];  // ds_load b64
        }
#pragma unroll
        for (int t = 0; t < NT; ++t) {
            acc[t] = __builtin_amdgcn_wmma_f32_16x16x4_f32(
                false, av, false, b[t], (short)0, acc[t], false, false);
        }
        a = an;
    }

    // epilogue: last k-step (k = K-4)
    {
        const int k = K - 4;
        v2f av = a;
        if (RELU) { av.x = fmaxf(av.x, 0.0f); av.y = fmaxf(av.y, 0.0f); }
        v2f b[NT];
#pragma unroll
        for (int t = 0; t < NT; ++t) {
            const float* bp = brow + (size_t)t * 16 * K + k;
            b[t].x = bp[0];
            b[t].y = bp[1];
        }
#pragma unroll
        for (int t = 0; t < NT; ++t) {
            acc[t] = __builtin_amdgcn_wmma_f32_16x16x4_f32(
                false, av, false, b[t], (short)0, acc[t], false, false);
        }
    }

    const int orow = wave * 16 + half * 8;
#pragma unroll
    for (int t = 0; t < NT; ++t) {
        const int ocol = t * 16 + l;
        if (ocol < C) {
#pragma unroll
            for (int r = 0; r < 8; ++r) {
                const int row = orow + r;
                if (row < N) H[(size_t)row * ldh + ocol] = acc[t][r];
            }
        }
    }
}

// ---------------- aggregation init: agg = bias + dinv^2 * h (self loop) ----------------

__global__ void k_init_agg(const float* __restrict__ h, const float* __restrict__ dinv,
                           const float* __restrict__ bias, float* __restrict__ agg,
                           int N, int C) {
    size_t idx = (size_t)blockIdx.x * blockDim.x + threadIdx.x;
    size_t total = (size_t)N * C;
    if (idx >= total) return;
    int i = (int)(idx / C);
    int c = (int)(idx - (size_t)i * C);
    float di = dinv[i];
    agg[idx] = bias[c] + di * di * h[idx];
}

// ---------------- edge scatter: agg[dst] += dinv[src]*dinv[dst] * h[src] ----------------

__global__ void k_scatter(const int* __restrict__ src, const int* __restrict__ dst,
                          const float* __restrict__ dinv, const float* __restrict__ h,
                          float* __restrict__ agg, int E, int C) {
    int e = blockIdx.x * blockDim.x + threadIdx.x;
    if (e >= E) return;
    int s = src[e];
    int d = dst[e];
    float coef = dinv[s] * dinv[d];
    const float* hp = h + (size_t)s * C;
    float* ap = agg + (size_t)d * C;
    if ((C & 3) == 0) {
        for (int c = 0; c < C; c += 4) {
            float4 v = *(const float4*)(hp + c);  // 16B aligned (C % 4 == 0)
            atomicAdd(ap + c + 0, coef * v.x);
            atomicAdd(ap + c + 1, coef * v.y);
            atomicAdd(ap + c + 2, coef * v.z);
            atomicAdd(ap + c + 3, coef * v.w);
        }
    } else {
        for (int c = 0; c < C; ++c) atomicAdd(ap + c, coef * hp[c]);
    }
}

// ---------------- pooling ----------------

__global__ void k_zero_pool(float* pooled, float* counts) {
    int t = threadIdx.x;
    if (t < NG * NC) pooled[t] = 0.0f;
    if (t < NG) counts[t] = 0.0f;
}

__global__ void k_pool(const float* __restrict__ h3, const int* __restrict__ batch,
                       float* __restrict__ pooled, float* __restrict__ counts, int N) {
    __shared__ float sp[NG * NC];
    __shared__ float sc[NG];
    int t = threadIdx.x;
    for (int j = t; j < NG * NC; j += blockDim.x) sp[j] = 0.0f;
    for (int j = t; j < NG; j += blockDim.x) sc[j] = 0.0f;
    __syncthreads();
    int i = blockIdx.x * blockDim.x + t;
    if (i < N) {
        int g = batch[i];
        atomicAdd(&sc[g], 1.0f);  // ds_add_f32
        const float* hp = h3 + (size_t)i * NC;
        for (int c = 0; c < NC; ++c) atomicAdd(&sp[g * NC + c], hp[c]);
    }
    __syncthreads();
    for (int j = t; j < NG * NC; j += blockDim.x)
        if (sp[j] != 0.0f) atomicAdd(&pooled[j], sp[j]);
    for (int j = t; j < NG; j += blockDim.x)
        if (sc[j] != 0.0f) atomicAdd(&counts[j], sc[j]);
}

__global__ void k_logsoftmax(const float* __restrict__ pooled,
                             const float* __restrict__ counts, float* __restrict__ out) {
    int g = threadIdx.x;
    if (g >= NG) return;
    float cnt = fmaxf(counts[g], 1.0f);
    float v[NC];
    float m = -3.0e38f;
    for (int c = 0; c < NC; ++c) {
        v[c] = pooled[g * NC + c] / cnt;
        m = fmaxf(m, v[c]);
    }
    float s = 0.0f;
    for (int c = 0; c < NC; ++c) s += expf(v[c] - m);
    float lse = m + logf(s);
    for (int c = 0; c < NC; ++c) out[g * NC + c] = v[c] - lse;
}

// ---------------- host-side launcher ----------------

extern "C" void kernel_launch(void* const* d_in, const int* in_sizes, int n_in,
                              void* d_out, int out_size, void* d_ws, size_t ws_size,
                              hipStream_t stream) {
    const float* x     = (const float*)d_in[0];
    const int*   ei    = (const int*)d_in[1];
    const int*   batch = (const int*)d_in[2];
    const float* W1 = (const float*)d_in[3];
    const float* b1 = (const float*)d_in[4];
    const float* W2 = (const float*)d_in[5];
    const float* b2 = (const float*)d_in[6];
    const float* W3 = (const float*)d_in[7];
    const float* b3 = (const float*)d_in[8];

    const int N = in_sizes[0] / IN_CH;
    const int E = in_sizes[1] / 2;
    const int* src = ei;       // edge_index[0]
    const int* dst = ei + E;   // edge_index[1]

    // workspace carve-out (floats)
    float* ws     = (float*)d_ws;
    float* dinv   = ws;                          // N
    float* bufA   = dinv + N;                    // N*HID   (gemm out)
    float* bufB   = bufA + (size_t)N * HID;      // N*HID   (aggregation)
    float* h3     = bufB + (size_t)N * HID;      // N*NC
    float* agg3   = h3 + (size_t)N * NC;         // N*NC
    float* pooled = agg3 + (size_t)N * NC;       // NG*NC
    float* counts = pooled + NG * NC;            // NG

    const int T = 256;
    dim3 blk(T);

    // normalization
    k_init_deg<<<(N + T - 1) / T, blk, 0, stream>>>(dinv, N);
    k_edge_deg<<<(E + T - 1) / T, blk, 0, stream>>>(dst, dinv, E);
    k_finalize_dinv<<<(N + T - 1) / T, blk, 0, stream>>>(dinv, N);

    const int tilesM = (N + 15) / 16;
    const int wavesPerBlk = T / 32;
    const int gemmBlocks = (tilesM + wavesPerBlk - 1) / wavesPerBlk;

    // ---- layer 1: h1 = x @ W1 ; agg1 = b1 + norm-scatter(h1) ----
    {
        k_gemm_wmma<4, false><<<gemmBlocks, blk, 4 * 16 * IN_CH * sizeof(float), stream>>>(
            x, W1, bufA, N, IN_CH, HID, HID);
        size_t tot = (size_t)N * HID;
        k_init_agg<<<(unsigned)((tot + T - 1) / T), blk, 0, stream>>>(bufA, dinv, b1, bufB, N, HID);
        k_scatter<<<(E + T - 1) / T, blk, 0, stream>>>(src, dst, dinv, bufA, bufB, E, HID);
    }

    // ---- layer 2: h2 = relu(agg1) @ W2 ; agg2 = b2 + norm-scatter(h2) ----
    {
        k_gemm_wmma<4, true><<<gemmBlocks, blk, 4 * 16 * HID * sizeof(float), stream>>>(
            bufB, W2, bufA, N, HID, HID, HID);
        size_t tot = (size_t)N * HID;
        k_init_agg<<<(unsigned)((tot + T - 1) / T), blk, 0, stream>>>(bufA, dinv, b2, bufB, N, HID);
        k_scatter<<<(E + T - 1) / T, blk, 0, stream>>>(src, dst, dinv, bufA, bufB, E, HID);
    }

    // ---- layer 3: h3 = relu(agg2) @ W3 ; agg3 = b3 + norm-scatter(h3) ----
    {
        k_gemm_wmma<1, true><<<gemmBlocks, blk, 1 * 16 * HID * sizeof(float), stream>>>(
            bufB, W3, h3, N, HID, NC, NC);
        size_t tot = (size_t)N * NC;
        k_init_agg<<<(unsigned)((tot + T - 1) / T), blk, 0, stream>>>(h3, dinv, b3, agg3, N, NC);
        k_scatter<<<(E + T - 1) / T, blk, 0, stream>>>(src, dst, dinv, h3, agg3, E, NC);
    }

    // ---- mean pool per graph + log_softmax ----
    k_zero_pool<<<1, blk, 0, stream>>>(pooled, counts);
    k_pool<<<(N + T - 1) / T, blk, 0, stream>>>(agg3, batch, pooled, counts, N);
    k_logsoftmax<<<1, 64, 0, stream>>>(pooled, counts, (float*)d_out);
}